// M2RAGNN_23252952940666
// MI455X (gfx1250) — compile-verified
//
#include <hip/hip_runtime.h>
#include <hip/hip_bf16.h>

typedef float v2f __attribute__((ext_vector_type(2)));
typedef float v8f __attribute__((ext_vector_type(8)));

#define HID 128
#define LRELU_SLOPE 0.2f
#define BN_EPS 1e-5f
#define MT 64           // GEMM output rows per block
#define SA_STRIDE 66    // LDS row stride (dwords): conflict-free, 8B-aligned pairs

static inline int cdiv_i(long long a, long long b) { return (int)((a + b - 1) / b); }

// =====================================================================
// WMMA GEMM: C[M,128] = A[M,K] @ B[K,128] (+ bias), all f32, row-major.
// One block: 64 rows x 128 cols. 8 waves; wave -> 16-col tile, 4 row-subtiles.
// Main K-loop is guard-free (full 64-wide chunks); ragged tail handled once.
// =====================================================================
__global__ __launch_bounds__(256) void m2r_gemm_wmma(
    const float* __restrict__ A, const float* __restrict__ B,
    const float* __restrict__ bias, float* __restrict__ C, int M, int K)
{
    __shared__ float sA[MT * SA_STRIDE];
    const int tid  = threadIdx.x;
    const int lane = tid & 31;
    const int wave = tid >> 5;        // 0..7 -> column tile n0 = wave*16
    const int m0   = blockIdx.x * MT;
    const int n0   = wave << 4;
    const int half = lane >> 4;       // 0: lanes 0-15, 1: lanes 16-31
    const int l16  = lane & 15;
    const int koff = half << 1;       // K sub-offset inside a 4-wide chunk

    v8f acc0 = {0.f,0.f,0.f,0.f,0.f,0.f,0.f,0.f};
    v8f acc1 = acc0, acc2 = acc0, acc3 = acc0;

    const float* Bcol = B + n0 + l16;
    const int aRow0 = (l16     ) * SA_STRIDE;
    const int aRow1 = (l16 + 16) * SA_STRIDE;
    const int aRow2 = (l16 + 32) * SA_STRIDE;
    const int aRow3 = (l16 + 48) * SA_STRIDE;

    int k0 = 0;
    // ---- main loop: full 64-wide K chunks, no K guards anywhere ----
    for (; k0 + 64 <= K; k0 += 64) {
        __syncthreads();
#pragma unroll
        for (int i = 0; i < 16; ++i) {          // 4096 floats / 256 threads
            int idx = tid + i * 256;
            int r = idx >> 6, kk = idx & 63;
            int m = m0 + r;
            sA[r * SA_STRIDE + kk] = (m < M) ? A[(size_t)m * K + k0 + kk] : 0.0f;
        }
        __syncthreads();
        const float* Bc = Bcol + (size_t)(k0 + koff) * HID;
#pragma unroll
        for (int ss = 0; ss < 16; ++ss) {
            v2f b;
            b.x = Bc[(size_t)(4 * ss    ) * HID];
            b.y = Bc[(size_t)(4 * ss + 1) * HID];
            const int ka = (ss << 2) + koff;
            v2f a;
            a.x = sA[aRow0 + ka]; a.y = sA[aRow0 + ka + 1];
            acc0 = __builtin_amdgcn_wmma_f32_16x16x4_f32(false, a, false, b, (short)0, acc0, false, false);
            a.x = sA[aRow1 + ka]; a.y = sA[aRow1 + ka + 1];
            acc1 = __builtin_amdgcn_wmma_f32_16x16x4_f32(false, a, false, b, (short)0, acc1, false, false);
            a.x = sA[aRow2 + ka]; a.y = sA[aRow2 + ka + 1];
            acc2 = __builtin_amdgcn_wmma_f32_16x16x4_f32(false, a, false, b, (short)0, acc2, false, false);
            a.x = sA[aRow3 + ka]; a.y = sA[aRow3 + ka + 1];
            acc3 = __builtin_amdgcn_wmma_f32_16x16x4_f32(false, a, false, b, (short)0, acc3, false, false);
        }
    }
    // ---- ragged K tail (runs at most once; K=2049 -> 1 col, K=5/1 -> all) ----
    if (k0 < K) {
        __syncthreads();
#pragma unroll
        for (int i = 0; i < 16; ++i) {
            int idx = tid + i * 256;
            int r = idx >> 6, kk = idx & 63;
            int m = m0 + r, k = k0 + kk;
            sA[r * SA_STRIDE + kk] = (m < M && k < K) ? A[(size_t)m * K + k] : 0.0f;
        }
        __syncthreads();
        const int steps = (K - k0 + 3) >> 2;
        for (int ss = 0; ss < steps; ++ss) {
            const int ka = (ss << 2) + koff;
            const int kg = k0 + ka;
            v2f b;
            b.x = (kg     < K) ? Bcol[(size_t)kg * HID]       : 0.0f;
            b.y = (kg + 1 < K) ? Bcol[(size_t)(kg + 1) * HID] : 0.0f;
            v2f a;
            a.x = sA[aRow0 + ka]; a.y = sA[aRow0 + ka + 1];
            acc0 = __builtin_amdgcn_wmma_f32_16x16x4_f32(false, a, false, b, (short)0, acc0, false, false);
            a.x = sA[aRow1 + ka]; a.y = sA[aRow1 + ka + 1];
            acc1 = __builtin_amdgcn_wmma_f32_16x16x4_f32(false, a, false, b, (short)0, acc1, false, false);
            a.x = sA[aRow2 + ka]; a.y = sA[aRow2 + ka + 1];
            acc2 = __builtin_amdgcn_wmma_f32_16x16x4_f32(false, a, false, b, (short)0, acc2, false, false);
            a.x = sA[aRow3 + ka]; a.y = sA[aRow3 + ka + 1];
            acc3 = __builtin_amdgcn_wmma_f32_16x16x4_f32(false, a, false, b, (short)0, acc3, false, false);
        }
    }
    // ---- epilogue: C layout VGPR r -> M = r (lanes 0-15) / r+8 (lanes 16-31) ----
    const int n  = n0 + l16;
    const float bv = bias ? bias[n] : 0.0f;
#pragma unroll
    for (int r = 0; r < 8; ++r) {
        int mb = m0 + r + (half << 3);
        if (mb      < M) C[(size_t)(mb     ) * HID + n] = acc0[r] + bv;
        if (mb + 16 < M) C[(size_t)(mb + 16) * HID + n] = acc1[r] + bv;
        if (mb + 32 < M) C[(size_t)(mb + 32) * HID + n] = acc2[r] + bv;
        if (mb + 48 < M) C[(size_t)(mb + 48) * HID + n] = acc3[r] + bv;
    }
}

// ---------------- per-node attention logits: out[n,h] = <hs[n,h,:], a[h,:]> ---
__global__ void m2r_rowdot64(const float* __restrict__ Hn, const float* __restrict__ av,
                             float* __restrict__ out, int N)
{
    int t = blockIdx.x * blockDim.x + threadIdx.x;
    if (t >= N * 2) return;
    int n = t >> 1, h = t & 1;
    const float* x = Hn + (size_t)n * HID + h * 64;
    const float* a = av + h * 64;
    float s = 0.f;
#pragma unroll
    for (int c = 0; c < 64; c += 4)
        s += x[c] * a[c] + x[c + 1] * a[c + 1] + x[c + 2] * a[c + 2] + x[c + 3] * a[c + 3];
    out[t] = s;
}

// ---------------- fills / accumulator init ----------------
__global__ void m2r_fill_f32(float* p, float v, size_t n) {
    size_t t = (size_t)blockIdx.x * blockDim.x + threadIdx.x;
    if (t < n) p[t] = v;
}
__global__ void m2r_fill_u32(unsigned* p, unsigned v, size_t n) {
    size_t t = (size_t)blockIdx.x * blockDim.x + threadIdx.x;
    if (t < n) p[t] = v;
}
__global__ void m2r_acc_init(float* acc, const float* __restrict__ b1,
                             const float* __restrict__ b2, size_t n) {
    size_t t = (size_t)blockIdx.x * blockDim.x + threadIdx.x;
    if (t < n) {
        int c = (int)(t & (HID - 1));
        acc[t] = b1[c] + (b2 ? b2[c] : 0.0f);
    }
}

// ---------------- monotonic float<->uint for atomicMax on f32 ----------------
__device__ __forceinline__ unsigned fenc(float x) {
    unsigned u = __float_as_uint(x);
    return (u & 0x80000000u) ? ~u : (u | 0x80000000u);
}
__device__ __forceinline__ float fdec(unsigned e) {
    unsigned u = (e & 0x80000000u) ? (e ^ 0x80000000u) : ~e;
    return __uint_as_float(u);
}
#define ENC_NEG_INF 0x007FFFFFu   // fenc(-inf)

__device__ __forceinline__ float lrelu(float v) { return v > 0.f ? v : LRELU_SLOPE * v; }

// ---------------- edge segment-max ----------------
__global__ void m2r_edge_max(const float* __restrict__ AS, const float* __restrict__ AD,
                             const int* __restrict__ row, const int* __restrict__ col,
                             unsigned* __restrict__ Menc, int E)
{
    int t = blockIdx.x * blockDim.x + threadIdx.x;
    if (t >= E * 2) return;
    int e = t >> 1, h = t & 1;
    int r = row[e], c = col[e];
    float v = lrelu(AS[r * 2 + h] + AD[c * 2 + h]);
    atomicMax(&Menc[c * 2 + h], fenc(v));
}
// decode + isfinite guard (empty segments -> 0, matching reference)
__global__ void m2r_seg_max_fix(const unsigned* __restrict__ Menc, float* __restrict__ Mf, int n)
{
    int t = blockIdx.x * blockDim.x + threadIdx.x;
    if (t >= n) return;
    float v = fdec(Menc[t]);
    if (!(v >= -3.0e38f && v <= 3.0e38f)) v = 0.0f;
    Mf[t] = v;
}

// ---------------- edge exp + denom ----------------
__global__ void m2r_edge_exp(const float* __restrict__ AS, const float* __restrict__ AD,
                             const int* __restrict__ row, const int* __restrict__ col,
                             const float* __restrict__ Mf, float* __restrict__ EX,
                             float* __restrict__ DEN, int E)
{
    int t = blockIdx.x * blockDim.x + threadIdx.x;
    if (t >= E * 2) return;
    int e = t >> 1, h = t & 1;
    int r = row[e], c = col[e];
    float v = lrelu(AS[r * 2 + h] + AD[c * 2 + h]);
    float ex = expf(v - Mf[c * 2 + h]);
    EX[t] = ex;
    atomicAdd(&DEN[c * 2 + h], ex);
}

// ---------------- weighted message scatter (L2-resident atomics) -------------
__global__ void m2r_edge_scatter(const float* __restrict__ HS, const int* __restrict__ row,
                                 const int* __restrict__ col, const float* __restrict__ EX,
                                 const float* __restrict__ DEN, float* __restrict__ ACC,
                                 long long total)
{
    long long t = (long long)blockIdx.x * blockDim.x + threadIdx.x;
    if (t >= total) return;
    int e = (int)(t >> 7);
    int ch = (int)(t & (HID - 1));
    int h = ch >> 6;
    int r = row[e], c = col[e];
    float alpha = EX[e * 2 + h] / (DEN[c * 2 + h] + 1e-16f);
    atomicAdd(&ACC[(size_t)c * HID + ch], HS[(size_t)r * HID + ch] * alpha);
}

// ---------------- BatchNorm over relu(x): per-channel stats ------------------
__global__ __launch_bounds__(256) void m2r_bnstat(const float* __restrict__ X,
                                                  float* __restrict__ sums, int N)
{
    int ch  = threadIdx.x & (HID - 1);
    int sub = threadIdx.x >> 7;   // 0/1: two rows per pass, fully coalesced
    float s = 0.f, s2 = 0.f;
    for (int n = blockIdx.x * 2 + sub; n < N; n += gridDim.x * 2) {
        float v = X[(size_t)n * HID + ch];
        v = v > 0.f ? v : 0.f;
        s += v; s2 += v * v;
    }
    __shared__ float red[256];
    red[threadIdx.x] = s;  __syncthreads();
    if (sub == 0) atomicAdd(&sums[ch], red[ch] + red[HID + ch]);
    __syncthreads();
    red[threadIdx.x] = s2; __syncthreads();
    if (sub == 0) atomicAdd(&sums[HID + ch], red[ch] + red[HID + ch]);
}
__global__ void m2r_bnapply(const float* __restrict__ X, const float* __restrict__ sums,
                            const float* __restrict__ gamma, const float* __restrict__ beta,
                            float* __restrict__ OUT, int N)
{
    size_t t = (size_t)blockIdx.x * blockDim.x + threadIdx.x;
    if (t >= (size_t)N * HID) return;
    int ch = (int)(t & (HID - 1));
    float mu  = sums[ch] / (float)N;
    float var = sums[HID + ch] / (float)N - mu * mu;   // population var (ddof=0)
    float v = X[t];
    v = v > 0.f ? v : 0.f;
    OUT[t] = gamma[ch] * (v - mu) * rsqrtf(var + BN_EPS) + beta[ch];
}

// ---------------- eqgat geometric scalar ----------------
__global__ __launch_bounds__(256) void m2r_geo_reduce(const float* __restrict__ ps,
                                                      const float* __restrict__ pd,
                                                      const int* __restrict__ row,
                                                      const int* __restrict__ col,
                                                      float* __restrict__ out, int E)
{
    __shared__ float red[256];
    int t = blockIdx.x * 256 + threadIdx.x;
    float d = 0.f;
    if (t < E) {
        int r = row[t], c = col[t];
        float dx = ps[r * 3 + 0] - pd[c * 3 + 0];
        float dy = ps[r * 3 + 1] - pd[c * 3 + 1];
        float dz = ps[r * 3 + 2] - pd[c * 3 + 2];
        d = sqrtf(dx * dx + dy * dy + dz * dz);
    }
    red[threadIdx.x] = d; __syncthreads();
    for (int s = 128; s > 0; s >>= 1) {
        if (threadIdx.x < s) red[threadIdx.x] += red[threadIdx.x + s];
        __syncthreads();
    }
    if (threadIdx.x == 0) atomicAdd(out, red[0]);
}
__global__ void m2r_shift(const float* __restrict__ X, const float* __restrict__ geo,
                          float scale, float* __restrict__ OUT, size_t n)
{
    size_t t = (size_t)blockIdx.x * blockDim.x + threadIdx.x;
    if (t < n) OUT[t] = X[t] + geo[0] * scale;
}

// ---------------- output heads ----------------
__global__ void m2r_head(const float* __restrict__ X, const float* __restrict__ W,
                         const float* __restrict__ b, float* __restrict__ out, int N)
{
    int n = blockIdx.x * blockDim.x + threadIdx.x;
    if (n >= N) return;
    const float* x = X + (size_t)n * HID;
    float s = 0.f;
#pragma unroll
    for (int c = 0; c < HID; ++c) s += x[c] * W[c];
    out[n] = s + b[0];
}

// =====================================================================
// Host orchestration
// =====================================================================
struct GatP { const float *Wsrc, *Wdst, *asrc, *adst, *bias; };

struct Scratch {
    float *HS, *HD, *AS, *AD, *Mf, *DEN, *EX;
    unsigned* Menc;
};

static void run_gat_rel(hipStream_t s,
                        const float* Xsrc, int Ns, const float* Xdst, int Nd,
                        const int* row, const int* col, int E,
                        const GatP& p, const Scratch& w, float* ACC)
{
    m2r_gemm_wmma<<<cdiv_i(Ns, MT), 256, 0, s>>>(Xsrc, p.Wsrc, nullptr, w.HS, Ns, HID);
    m2r_gemm_wmma<<<cdiv_i(Nd, MT), 256, 0, s>>>(Xdst, p.Wdst, nullptr, w.HD, Nd, HID);
    m2r_rowdot64<<<cdiv_i((long long)Ns * 2, 256), 256, 0, s>>>(w.HS, p.asrc, w.AS, Ns);
    m2r_rowdot64<<<cdiv_i((long long)Nd * 2, 256), 256, 0, s>>>(w.HD, p.adst, w.AD, Nd);
    m2r_fill_u32<<<cdiv_i((long long)Nd * 2, 256), 256, 0, s>>>(w.Menc, ENC_NEG_INF, (size_t)Nd * 2);
    m2r_edge_max<<<cdiv_i((long long)E * 2, 256), 256, 0, s>>>(w.AS, w.AD, row, col, w.Menc, E);
    m2r_seg_max_fix<<<cdiv_i((long long)Nd * 2, 256), 256, 0, s>>>(w.Menc, w.Mf, Nd * 2);
    m2r_fill_f32<<<cdiv_i((long long)Nd * 2, 256), 256, 0, s>>>(w.DEN, 0.0f, (size_t)Nd * 2);
    m2r_edge_exp<<<cdiv_i((long long)E * 2, 256), 256, 0, s>>>(w.AS, w.AD, row, col, w.Mf, w.EX, w.DEN, E);
    long long total = (long long)E * HID;
    m2r_edge_scatter<<<cdiv_i(total, 256), 256, 0, s>>>(w.HS, row, col, w.EX, w.DEN, ACC, total);
}

static void run_bn(hipStream_t s, const float* X, int N, const float* gamma,
                   const float* beta, float* sums, float* OUT)
{
    m2r_fill_f32<<<1, 256, 0, s>>>(sums, 0.0f, 256);
    m2r_bnstat<<<256, 256, 0, s>>>(X, sums, N);
    m2r_bnapply<<<cdiv_i((long long)N * HID, 256), 256, 0, s>>>(X, sums, gamma, beta, OUT, N);
}

extern "C" void kernel_launch(void* const* d_in, const int* in_sizes, int n_in,
                              void* d_out, int out_size, void* d_ws, size_t ws_size,
                              hipStream_t stream)
{
    hipStream_t s = stream;
    const float* XPREC = (const float*)d_in[0];
    const float* XPROD = (const float*)d_in[1];
    const float* XCOND = (const float*)d_in[2];
    const float* XTGT  = (const float*)d_in[3];
    const float* POSP  = (const float*)d_in[4];
    const float* POST  = (const float*)d_in[5];
    const int *E1S = (const int*)d_in[6],  *E1D = (const int*)d_in[7];
    const int *E2S = (const int*)d_in[8],  *E2D = (const int*)d_in[9];
    const int *E3S = (const int*)d_in[10], *E3D = (const int*)d_in[11];

    const int Np  = in_sizes[0] / 2049;
    const int Npr = in_sizes[1] / 2049;
    const int Nc  = in_sizes[2] / 5;
    const int Nt  = in_sizes[3];
    const int E1 = in_sizes[6], E2 = in_sizes[8], E3 = in_sizes[10];

#define FP(i) ((const float*)d_in[(i)])
    const float *PW_PREC = FP(12), *PB_PREC = FP(13);
    const float *PW_PROD = FP(14), *PB_PROD = FP(15);
    const float *PW_COND = FP(16), *PB_COND = FP(17);
    const float *PW_TGT  = FP(18), *PB_TGT  = FP(19);
    auto gat_at = [&](int b) -> GatP {
        return GatP{FP(b), FP(b + 1), FP(b + 2), FP(b + 3), FP(b + 4)};
    };
    GatP c1r1 = gat_at(20), c1r2 = gat_at(25);
    GatP c1r3 = gat_at(30); const float *C1R3_PW = FP(35), *C1R3_PB = FP(36);
    GatP c2r1 = gat_at(37), c2r2 = gat_at(42);
    GatP c2r3 = gat_at(47); const float *C2R3_PW = FP(52), *C2R3_PB = FP(53);
    const float *BN_G = FP(54), *BN_B = FP(55);
    const float *YW = FP(56), *YB = FP(57), *AW = FP(58), *AB = FP(59);
#undef FP

    // ---- workspace layout (floats) ----
    float* W = (float*)d_ws;
    size_t o = 0;
    auto take = [&](size_t n) { float* p = W + o; o += n; return p; };
    const int maxNs = Np > Npr ? (Np > Nc ? Np : Nc) : (Npr > Nc ? Npr : Nc);
    const int maxNd = Npr > Nt ? Npr : Nt;
    const int maxE  = E1 > E2 ? (E1 > E3 ? E1 : E3) : (E2 > E3 ? E2 : E3);

    float* H_PREC   = take((size_t)Np  * HID);
    float* H_PROD   = take((size_t)Npr * HID);
    float* H_COND   = take((size_t)Nc  * HID);
    float* H_TGT    = take((size_t)Nt  * HID);
    float* PROD_CUR = take((size_t)Npr * HID);
    float* TGT_CUR  = take((size_t)Nt  * HID);
    float* ACC      = take((size_t)Npr * HID);
    float* XSHIFT   = take((size_t)Npr * HID);
    float* TGT_ACC  = take((size_t)Nt  * HID);
    float* TGT_RAW  = take((size_t)Nt  * HID);
    Scratch w;
    w.HS   = take((size_t)maxNs * HID);
    w.HD   = take((size_t)maxNd * HID);
    w.AS   = take((size_t)maxNs * 2);
    w.AD   = take((size_t)maxNd * 2);
    w.Menc = (unsigned*)take((size_t)maxNd * 2);
    w.Mf   = take((size_t)maxNd * 2);
    w.DEN  = take((size_t)maxNd * 2);
    w.EX   = take((size_t)maxE * 2);
    float* SUMS = take(256);
    float* GEO  = take(1);
    (void)ws_size; (void)n_in; (void)out_size;

    float* OUT_Y = (float*)d_out;   // pred_yield [Npr]
    float* OUT_A = OUT_Y + Npr;     // pred_activity [Nt]

    // ---- phase 0: input projections (WMMA) ----
    m2r_gemm_wmma<<<cdiv_i(Np, MT),  256, 0, s>>>(XPREC, PW_PREC, PB_PREC, H_PREC, Np, 2049);
    m2r_gemm_wmma<<<cdiv_i(Npr, MT), 256, 0, s>>>(XPROD, PW_PROD, PB_PROD, H_PROD, Npr, 2049);
    m2r_gemm_wmma<<<cdiv_i(Nc, MT),  256, 0, s>>>(XCOND, PW_COND, PB_COND, H_COND, Nc, 5);
    m2r_gemm_wmma<<<cdiv_i(Nt, MT),  256, 0, s>>>(XTGT,  PW_TGT,  PB_TGT,  H_TGT,  Nt, 1);

    // geometric scalar (positions/edges fixed -> shared by both layers)
    m2r_fill_f32<<<1, 256, 0, s>>>(GEO, 0.0f, 1);
    m2r_geo_reduce<<<cdiv_i(E3, 256), 256, 0, s>>>(POSP, POST, E3S, E3D, GEO, E3);
    const float geoScale = 0.1f / (float)E3;

    // ---- layer 1: products (sum of two relations into one accumulator) ----
    m2r_acc_init<<<cdiv_i((long long)Npr * HID, 256), 256, 0, s>>>(ACC, c1r1.bias, c1r2.bias, (size_t)Npr * HID);
    run_gat_rel(s, H_PREC, Np, H_PROD, Npr, E1S, E1D, E1, c1r1, w, ACC);
    run_gat_rel(s, H_COND, Nc, H_PROD, Npr, E2S, E2D, E2, c1r2, w, ACC);
    run_bn(s, ACC, Npr, BN_G, BN_B, SUMS, PROD_CUR);                 // prod1

    // ---- layer 1: targets (eqgat) ----
    m2r_shift<<<cdiv_i((long long)Npr * HID, 256), 256, 0, s>>>(H_PROD, GEO, geoScale, XSHIFT, (size_t)Npr * HID);
    m2r_acc_init<<<cdiv_i((long long)Nt * HID, 256), 256, 0, s>>>(TGT_ACC, c1r3.bias, nullptr, (size_t)Nt * HID);
    run_gat_rel(s, XSHIFT, Npr, H_TGT, Nt, E3S, E3D, E3, c1r3, w, TGT_ACC);
    m2r_gemm_wmma<<<cdiv_i(Nt, MT), 256, 0, s>>>(TGT_ACC, C1R3_PW, C1R3_PB, TGT_RAW, Nt, HID);
    run_bn(s, TGT_RAW, Nt, BN_G, BN_B, SUMS, TGT_CUR);               // tgt1

    // ---- layer 2: products (result parked in ACC until targets consume prod1) ----
    m2r_acc_init<<<cdiv_i((long long)Npr * HID, 256), 256, 0, s>>>(ACC, c2r1.bias, c2r2.bias, (size_t)Npr * HID);
    run_gat_rel(s, H_PREC, Np, PROD_CUR, Npr, E1S, E1D, E1, c2r1, w, ACC);
    run_gat_rel(s, H_COND, Nc, PROD_CUR, Npr, E2S, E2D, E2, c2r2, w, ACC);

    // ---- layer 2: targets (reads prod1/tgt1 before they are overwritten) ----
    m2r_shift<<<cdiv_i((long long)Npr * HID, 256), 256, 0, s>>>(PROD_CUR, GEO, geoScale, XSHIFT, (size_t)Npr * HID);
    m2r_acc_init<<<cdiv_i((long long)Nt * HID, 256), 256, 0, s>>>(TGT_ACC, c2r3.bias, nullptr, (size_t)Nt * HID);
    run_gat_rel(s, XSHIFT, Npr, TGT_CUR, Nt, E3S, E3D, E3, c2r3, w, TGT_ACC);
    m2r_gemm_wmma<<<cdiv_i(Nt, MT), 256, 0, s>>>(TGT_ACC, C2R3_PW, C2R3_PB, TGT_RAW, Nt, HID);

    run_bn(s, ACC, Npr, BN_G, BN_B, SUMS, PROD_CUR);                 // prod2
    run_bn(s, TGT_RAW, Nt, BN_G, BN_B, SUMS, TGT_CUR);               // tgt2

    // ---- heads ----
    m2r_head<<<cdiv_i(Npr, 256), 256, 0, s>>>(PROD_CUR, YW, YB, OUT_Y, Npr);
    m2r_head<<<cdiv_i(Nt, 256),  256, 0, s>>>(TGT_CUR,  AW, AB, OUT_A, Nt);
}